// TopKRouter_50843822850155
// MI455X (gfx1250) — compile-verified
//
#include <hip/hip_runtime.h>

// ---------------------------------------------------------------------------
// MoE Top-2 router for MI455X (gfx1250, wave32).
//   logits = HS[M,K] x W[K,E]   (fp32, via V_WMMA_F32_16X16X4_F32)
//   probs  = softmax(logits)    -> written out
//   top2 (renormalized) + indices
//   aux    = sum_e( mean_tok(probs_e) * log(mean_tok(probs_e)+1e-9) )
//   counts = zeros
// ---------------------------------------------------------------------------

typedef __attribute__((ext_vector_type(2))) float v2f;
typedef __attribute__((ext_vector_type(4))) float v4f;
typedef __attribute__((ext_vector_type(8))) float v8f;

#define HDIM 2048
#define NEXP 64
#define ROWS_PB 32   // tokens per workgroup (one wave32)

__global__ __launch_bounds__(32) void moe_router_main(
    const float* __restrict__ hs,        // [M, HDIM]
    const float* __restrict__ w,         // [HDIM, NEXP]
    float* __restrict__ topk_p,          // [M, 2]
    int*   __restrict__ topk_i,          // [M, 2]
    float* __restrict__ probs_out,       // [M, NEXP]
    float* __restrict__ expert_sums)     // [NEXP] (global fp32 accum)
{
    __shared__ float lgt[ROWS_PB][NEXP + 1];   // +1 pad -> conflict-free columns

    const int lane = threadIdx.x;   // 0..31
    const int hi   = lane >> 4;     // 0: K=0,1 ; 1: K=2,3  (WMMA A layout)
    const int lm   = lane & 15;
    const int row0 = blockIdx.x * ROWS_PB;

    v8f acc[2][4] = {};             // 2 M-strips x 4 N-tiles of 16x16 fp32

    // Both lane halves load the SAME row (broadcast-coalesced b128 loads);
    // halves keep different K-pairs to match the 16x4 fp32 A layout.
    const float* aptr0 = hs + (size_t)(row0 + lm) * HDIM;
    const float* aptr1 = aptr0 + (size_t)16 * HDIM;
    // B layout: VGPR j, lanes<16 -> B[k+j][col], lanes>=16 -> B[k+2+j][col]
    const float* bbase = w + (size_t)(2 * hi) * NEXP + lm;

    #pragma unroll 2
    for (int k = 0; k < HDIM; k += 4) {
        v4f a0 = *(const v4f*)(aptr0 + k);
        v4f a1 = *(const v4f*)(aptr1 + k);
        v2f af0, af1;
        af0.x = hi ? a0.z : a0.x;  af0.y = hi ? a0.w : a0.y;
        af1.x = hi ? a1.z : a1.x;  af1.y = hi ? a1.w : a1.y;

        const float* brow = bbase + (size_t)k * NEXP;
        #pragma unroll
        for (int t = 0; t < 4; ++t) {
            v2f bf;
            bf.x = brow[16 * t];            // row k + 2*hi
            bf.y = brow[NEXP + 16 * t];     // row k + 2*hi + 1
            acc[0][t] = __builtin_amdgcn_wmma_f32_16x16x4_f32(
                false, af0, false, bf, (short)0, acc[0][t], false, false);
            acc[1][t] = __builtin_amdgcn_wmma_f32_16x16x4_f32(
                false, af1, false, bf, (short)0, acc[1][t], false, false);
        }
    }

    // Spill logits to LDS: C layout: VGPR v, lanes<16 -> M=v, lanes>=16 -> M=v+8
    #pragma unroll
    for (int s = 0; s < 2; ++s)
        #pragma unroll
        for (int t = 0; t < 4; ++t)
            #pragma unroll
            for (int v = 0; v < 8; ++v)
                lgt[16 * s + 8 * hi + v][16 * t + lm] = acc[s][t][v];

    __syncthreads();

    // One lane per token: softmax over 64 experts, write probs, find top-2.
    {
        float* rowp = lgt[lane];
        float m = rowp[0];
        #pragma unroll 8
        for (int e = 1; e < NEXP; ++e) m = fmaxf(m, rowp[e]);
        float sum = 0.0f;
        #pragma unroll 8
        for (int e = 0; e < NEXP; ++e) {
            float ex = __expf(rowp[e] - m);
            rowp[e] = ex;
            sum += ex;
        }
        const float inv = 1.0f / sum;

        float p0 = -1.0f, p1 = -1.0f;
        int   i0 = 0,     i1 = 0;
        float* po = probs_out + (size_t)(row0 + lane) * NEXP; // odd base offset -> scalar b32 stores
        #pragma unroll 8
        for (int e = 0; e < NEXP; ++e) {
            float p = rowp[e] * inv;
            rowp[e] = p;          // keep normalized probs for column reduction
            po[e]   = p;
            if (p > p0)      { p1 = p0; i1 = i0; p0 = p; i0 = e; }
            else if (p > p1) { p1 = p;  i1 = e; }
        }
        const float rn = 1.0f / (p0 + p1);
        const size_t ob = (size_t)(row0 + lane) * 2;
        topk_p[ob]     = p0 * rn;
        topk_p[ob + 1] = p1 * rn;
        topk_i[ob]     = i0;
        topk_i[ob + 1] = i1;
    }

    __syncthreads();

    // Per-expert partial sums for this 32-token block -> one atomic per expert.
    #pragma unroll
    for (int cc = 0; cc < 2; ++cc) {
        const int c = lane + 32 * cc;
        float s = 0.0f;
        #pragma unroll 8
        for (int r = 0; r < ROWS_PB; ++r) s += lgt[r][c];
        atomicAdd(&expert_sums[c], s);
    }
}

__global__ __launch_bounds__(64) void moe_router_zero_ws(float* __restrict__ p)
{
    p[threadIdx.x] = 0.0f;
}

__global__ __launch_bounds__(64) void moe_router_finalize(
    const float* __restrict__ expert_sums,
    float* __restrict__ aux_out,
    int*   __restrict__ counts_out,
    float inv_tokens)
{
    __shared__ float red[NEXP];
    const int t = threadIdx.x;
    const float load = expert_sums[t] * inv_tokens;
    red[t] = load * __logf(load + 1e-9f);   // mean_e(...)*NEXP == sum_e(...)
    counts_out[t] = 0;
    __syncthreads();
    #pragma unroll
    for (int off = 32; off > 0; off >>= 1) {
        if (t < off) red[t] += red[t + off];
        __syncthreads();
    }
    if (t == 0) aux_out[0] = red[0];
}

extern "C" void kernel_launch(void* const* d_in, const int* in_sizes, int n_in,
                              void* d_out, int out_size, void* d_ws, size_t ws_size,
                              hipStream_t stream)
{
    const float* hs = (const float*)d_in[0];   // [B,S,H] -> [M, 2048]
    const float* w  = (const float*)d_in[1];   // [2048, 64]
    const int M = in_sizes[0] / HDIM;          // 16384

    // d_out layout (flat, return order):
    //   top_k_probs [M,2] | top_k_indices [M,2] (int bits) | aux (1) |
    //   expert_counts [64] (int bits) | routing_probs [M,64]
    float* out     = (float*)d_out;
    float* topk_p  = out;
    int*   topk_i  = (int*)(out + (size_t)M * 2);
    float* aux     = out + (size_t)M * 4;
    int*   counts  = (int*)(out + (size_t)M * 4 + 1);
    float* probs   = out + (size_t)M * 4 + 1 + NEXP;

    float* esums = (float*)d_ws;               // [64] fp32 scratch

    moe_router_zero_ws<<<1, NEXP, 0, stream>>>(esums);
    moe_router_main<<<M / ROWS_PB, 32, 0, stream>>>(hs, w, topk_p, topk_i, probs, esums);
    moe_router_finalize<<<1, NEXP, 0, stream>>>(esums, aux, counts, 1.0f / (float)M);
}